// EMDLoss_64338610094789
// MI455X (gfx1250) — compile-verified
//
#include <hip/hip_runtime.h>

// ---------------------------------------------------------------------------
// Approx-EMD (Fan et al. approxmatch) fully fused for gfx1250 (MI455X).
//  b=16, n=m=2048, 3-D points.  No n*m matrix is ever materialized:
//  d2 tiles are recomputed on the fly with V_WMMA_F32_16X16X4_F32.
//  Levels arrive pre-multiplied by log2(e); exponentials use v_exp_f32
//  directly via __builtin_amdgcn_exp2f (fma -> exp, no extra scaling muls).
// ---------------------------------------------------------------------------

typedef __attribute__((ext_vector_type(2))) float v2f;
typedef __attribute__((ext_vector_type(8))) float v8f;

#define BATCH 16
#define NPTS  2048
#define MPTS  2048
#define NT    (NPTS / 16)   // 128 row tiles
#define MT    (MPTS / 16)   // 128 col tiles

__device__ __forceinline__ float lane_get(float v, int src) {
    return __shfl(v, src, 32);
}

// Reduce within each 16-lane half (lanes 0-15 and 16-31 independently).
__device__ __forceinline__ float half_reduce(float v) {
    v += __shfl_xor(v, 1, 32);
    v += __shfl_xor(v, 2, 32);
    v += __shfl_xor(v, 4, 32);
    v += __shfl_xor(v, 8, 32);
    return v;
}

__device__ __forceinline__ v8f wmma_tile(v2f a, v2f b) {
    v8f c = {};
    c = __builtin_amdgcn_wmma_f32_16x16x4_f32(
            /*neg_a=*/false, a, /*neg_b=*/false, b,
            /*c_mod=*/(short)0, c, /*reuse_a=*/false, /*reuse_b=*/false);
    return c;
}

// ---------------------------------------------------------------------------
__global__ void emd_init(float* __restrict__ remainl,
                         float* __restrict__ remainr,
                         float* __restrict__ cost) {
    int t = blockIdx.x * blockDim.x + threadIdx.x;
    if (t < BATCH * NPTS) remainl[t] = 1.0f;   // factorl = max(n,m)/n = 1
    if (t < BATCH * MPTS) remainr[t] = 1.0f;   // factorr = 1
    if (t < BATCH)        cost[t]    = 0.0f;
}

// ---------------------------------------------------------------------------
// Pass 1 (row-parallel): rfac_i = remainl_i / (sum_j exp(lvl*d2)*remainr_j + eps)
//   WMMA: A_i=(-2x,1), B_j=(y,|y|^2) -> D = -2x.y + |y_j|^2 ; exponent adds
//   lvl2*|x_i|^2 (per-strip constant ls1[v]).
__global__ void emd_rowpass(const float* __restrict__ xyz1,
                            const float* __restrict__ xyz2,
                            const float* __restrict__ remainl,
                            const float* __restrict__ remainr,
                            float* __restrict__ rfac,
                            float lvl2) {
    int wave  = (blockIdx.x * blockDim.x + threadIdx.x) >> 5;
    int lane  = threadIdx.x & 31;
    int half  = lane >> 4;
    int l16   = lane & 15;
    int batch = wave / NT;
    int ibase = (wave % NT) * 16;

    const float* x = xyz1 + ((size_t)batch * NPTS + ibase + l16) * 3;
    float x0 = x[0], x1 = x[1], x2 = x[2];
    float sq1 = x0 * x0 + x1 * x1 + x2 * x2;
    v2f a;
    a.x = half ? (-2.0f * x2) : (-2.0f * x0);
    a.y = half ? 1.0f         : (-2.0f * x1);

    float ls1[8];
#pragma unroll
    for (int v = 0; v < 8; ++v) ls1[v] = lvl2 * lane_get(sq1, v + 8 * half);

    float racc[8];
#pragma unroll
    for (int v = 0; v < 8; ++v) racc[v] = 0.0f;

    const float* ybase = xyz2 + (size_t)batch * MPTS * 3;
    const float* rrb   = remainr + (size_t)batch * MPTS;

    for (int jt = 0; jt < MT; ++jt) {
        int j = jt * 16 + l16;
        const float* y = ybase + (size_t)j * 3;
        float y0 = y[0], y1 = y[1], y2 = y[2];
        float sq2 = y0 * y0 + y1 * y1 + y2 * y2;
        v2f b;
        b.x = half ? y2  : y0;
        b.y = half ? sq2 : y1;
        float rr = rrb[j];

        v8f c = wmma_tile(a, b);
#pragma unroll
        for (int v = 0; v < 8; ++v) {
            float e = __builtin_amdgcn_exp2f(__builtin_fmaf(lvl2, c[v], ls1[v]));
            racc[v] = __builtin_fmaf(e, rr, racc[v]);
        }
    }

#pragma unroll
    for (int v = 0; v < 8; ++v) racc[v] = half_reduce(racc[v]);

    if (l16 == 0) {
#pragma unroll
        for (int v = 0; v < 8; ++v) {
            size_t idx = (size_t)batch * NPTS + ibase + v + 8 * half;
            rfac[idx] = remainl[idx] / (racc[v] + 1e-9f);
        }
    }
}

// ---------------------------------------------------------------------------
// Pass 2 (column-parallel): t_j = sum_i exp(lvl*d2)*rfac_i ;
//   ss2_j = remainr_j * t_j ; s_j = min(remainr_j/(ss2_j+eps), 1)
//   remainr_next_j = max(remainr_j - s_j*ss2_j, 0)
//   WMMA fold swapped: A_i=(-2x,|x_i|^2), B_j=(y,1) -> D = -2x.y + |x_i|^2 ;
//   exponent adds lvl2*|y_j|^2 (per-strip, per-lane constant — no shuffles).
__global__ void emd_colpass(const float* __restrict__ xyz1,
                            const float* __restrict__ xyz2,
                            const float* __restrict__ rfac,
                            const float* __restrict__ remainr,
                            float* __restrict__ s_out,
                            float* __restrict__ remainr_next,
                            float lvl2) {
    int wave  = (blockIdx.x * blockDim.x + threadIdx.x) >> 5;
    int lane  = threadIdx.x & 31;
    int half  = lane >> 4;
    int l16   = lane & 15;
    int batch = wave / MT;
    int jbase = (wave % MT) * 16;

    const float* y = xyz2 + ((size_t)batch * MPTS + jbase + l16) * 3;
    float y0 = y[0], y1 = y[1], y2 = y[2];
    float lsq2 = lvl2 * (y0 * y0 + y1 * y1 + y2 * y2);
    v2f b;
    b.x = half ? y2   : y0;
    b.y = half ? 1.0f : y1;

    const float* xbase = xyz1 + (size_t)batch * NPTS * 3;
    const float* rfb   = rfac + (size_t)batch * NPTS;

    float cacc = 0.0f;
    for (int it = 0; it < NT; ++it) {
        int i = it * 16 + l16;
        const float* x = xbase + (size_t)i * 3;
        float x0 = x[0], x1 = x[1], x2 = x[2];
        float sq1 = x0 * x0 + x1 * x1 + x2 * x2;
        float rf  = rfb[i];
        v2f a;
        a.x = half ? (-2.0f * x2) : (-2.0f * x0);
        a.y = half ? sq1          : (-2.0f * x1);

        float rf8[8];
#pragma unroll
        for (int v = 0; v < 8; ++v) rf8[v] = lane_get(rf, v + 8 * half);

        v8f c = wmma_tile(a, b);
#pragma unroll
        for (int v = 0; v < 8; ++v) {
            float e = __builtin_amdgcn_exp2f(__builtin_fmaf(lvl2, c[v], lsq2));
            cacc = __builtin_fmaf(e, rf8[v], cacc);
        }
    }

    // column j total = lanes j and j+16
    float t = cacc + __shfl_xor(cacc, 16, 32);
    if (half == 0) {
        size_t idx = (size_t)batch * MPTS + jbase + l16;
        float rr  = remainr[idx];
        float ss2 = rr * t;
        float s   = fminf(rr / (ss2 + 1e-9f), 1.0f);
        s_out[idx]        = s;
        remainr_next[idx] = fmaxf(rr - s * ss2, 0.0f);
    }
}

// ---------------------------------------------------------------------------
// Pass 3 (row-parallel): w_ij = e_ij*remainr_j*s_j*rfac_i
//   cost[b] += sum w*sqrt(d2);  remainl_i -= sum_j w_ij
__global__ void emd_rowupdate(const float* __restrict__ xyz1,
                              const float* __restrict__ xyz2,
                              const float* __restrict__ rfac,
                              const float* __restrict__ remainr,
                              const float* __restrict__ s_in,
                              float* __restrict__ remainl,
                              float* __restrict__ cost,
                              float lvl2) {
    int wave  = (blockIdx.x * blockDim.x + threadIdx.x) >> 5;
    int lane  = threadIdx.x & 31;
    int half  = lane >> 4;
    int l16   = lane & 15;
    int batch = wave / NT;
    int ibase = (wave % NT) * 16;

    const float* x = xyz1 + ((size_t)batch * NPTS + ibase + l16) * 3;
    float x0 = x[0], x1 = x[1], x2 = x[2];
    float sq1 = x0 * x0 + x1 * x1 + x2 * x2;
    float rf  = rfac[(size_t)batch * NPTS + ibase + l16];
    v2f a;
    a.x = half ? (-2.0f * x2) : (-2.0f * x0);
    a.y = half ? 1.0f         : (-2.0f * x1);

    float s1[8], ls1[8], rf8[8];
#pragma unroll
    for (int v = 0; v < 8; ++v) {
        s1[v]  = lane_get(sq1, v + 8 * half);
        ls1[v] = lvl2 * s1[v];
        rf8[v] = lane_get(rf, v + 8 * half);
    }

    float racc[8];
#pragma unroll
    for (int v = 0; v < 8; ++v) racc[v] = 0.0f;
    float costacc = 0.0f;

    const float* ybase = xyz2 + (size_t)batch * MPTS * 3;
    const float* rrb   = remainr + (size_t)batch * MPTS;
    const float* sb    = s_in   + (size_t)batch * MPTS;

    for (int jt = 0; jt < MT; ++jt) {
        int j = jt * 16 + l16;
        const float* y = ybase + (size_t)j * 3;
        float y0 = y[0], y1 = y[1], y2 = y[2];
        float sq2 = y0 * y0 + y1 * y1 + y2 * y2;
        v2f b;
        b.x = half ? y2  : y0;
        b.y = half ? sq2 : y1;
        float rq = rrb[j] * sb[j];

        v8f c = wmma_tile(a, b);
#pragma unroll
        for (int v = 0; v < 8; ++v) {
            float d2 = c[v] + s1[v];
            float e  = __builtin_amdgcn_exp2f(__builtin_fmaf(lvl2, c[v], ls1[v]));
            float t  = e * rq;
            racc[v] += t;
            float dist = __builtin_amdgcn_sqrtf(fmaxf(d2, 1e-12f));
            costacc = __builtin_fmaf(t * rf8[v], dist, costacc);
        }
    }

#pragma unroll
    for (int v = 0; v < 8; ++v) racc[v] = half_reduce(racc[v]);

    if (l16 == 0) {
#pragma unroll
        for (int v = 0; v < 8; ++v) {
            size_t idx = (size_t)batch * NPTS + ibase + v + 8 * half;
            remainl[idx] = fmaxf(remainl[idx] - rf8[v] * racc[v], 0.0f);
        }
    }

    // full-wave cost reduction
    costacc += __shfl_xor(costacc, 1, 32);
    costacc += __shfl_xor(costacc, 2, 32);
    costacc += __shfl_xor(costacc, 4, 32);
    costacc += __shfl_xor(costacc, 8, 32);
    costacc += __shfl_xor(costacc, 16, 32);
    if (lane == 0) atomicAdd(&cost[batch], costacc);
}

// ---------------------------------------------------------------------------
extern "C" void kernel_launch(void* const* d_in, const int* in_sizes, int n_in,
                              void* d_out, int out_size, void* d_ws, size_t ws_size,
                              hipStream_t stream) {
    const float* xyz1 = (const float*)d_in[0];
    const float* xyz2 = (const float*)d_in[1];
    float* cost = (float*)d_out;

    float* ws      = (float*)d_ws;
    float* remainl = ws;                                   // BATCH*NPTS
    float* rfac    = ws + (size_t)BATCH * NPTS;            // BATCH*NPTS
    float* s_arr   = ws + (size_t)2 * BATCH * NPTS;        // BATCH*MPTS
    float* remr0   = s_arr + (size_t)BATCH * MPTS;         // BATCH*MPTS
    float* remr1   = remr0 + (size_t)BATCH * MPTS;         // BATCH*MPTS

    dim3 blk(256);
    emd_init<<<(BATCH * NPTS + 255) / 256, blk, 0, stream>>>(remainl, remr0, cost);

    // levels pre-multiplied by log2(e) so device uses v_exp_f32 (exp2) directly
    const float log2e = 1.44269504088896340736f;
    static const float levels[10] = {
        -16384.0f, -4096.0f, -1024.0f, -256.0f, -64.0f,
        -16.0f,    -4.0f,    -1.0f,    -0.25f,  0.0f};

    int nwaves  = BATCH * NT;          // 2048 waves (one per 16-row/col strip)
    int nblocks = nwaves * 32 / 256;   // 256 blocks of 8 wave32s

    float* cur = remr0;
    float* nxt = remr1;
    for (int r = 0; r < 10; ++r) {
        float lv2 = levels[r] * log2e;
        emd_rowpass  <<<nblocks, blk, 0, stream>>>(xyz1, xyz2, remainl, cur, rfac, lv2);
        emd_colpass  <<<nblocks, blk, 0, stream>>>(xyz1, xyz2, rfac, cur, s_arr, nxt, lv2);
        emd_rowupdate<<<nblocks, blk, 0, stream>>>(xyz1, xyz2, rfac, cur, s_arr, remainl, cost, lv2);
        float* t = cur; cur = nxt; nxt = t;
    }
}